// Neuralnetwork_52175262712507
// MI455X (gfx1250) — compile-verified
//
#include <hip/hip_runtime.h>

// Elementwise "Duffing-map + soft window" kernel for MI455X (gfx1250).
// Memory-bound: 12 B/point, 23.3 TB/s HBM -> ~8.6 us floor for B = 16.7M.
// Strategy: branch-free fast kernel, 16 points/thread as 8 float2 chains so
// the recurrence lowers to v_pk_fma_f32 (CDNA5 double-rate packed fp32, with
// SGPR-broadcast constants), 8x b128 loads + 4x NT b128 stores per thread,
// gfx1250 v_tanh_f32 on the TRANS pipe (co-executes with VALU).

typedef float v4f __attribute__((ext_vector_type(4)));
typedef float v2f __attribute__((ext_vector_type(2)));

#define PTS_PER_THREAD 16
#define THREADS 256
#define PTS_PER_BLOCK (PTS_PER_THREAD * THREADS)   // 4096

#if defined(__AMDGCN__) && __has_builtin(__builtin_amdgcn_tanhf)
__device__ __forceinline__ float fast_tanh(float x) {
    return __builtin_amdgcn_tanhf(x);   // gfx1250 v_tanh_f32 (TRANS pipe)
}
#else
__device__ __forceinline__ float fast_tanh(float x) { return tanhf(x); }
#endif

__device__ __forceinline__ v2f splat2(float s) {
    v2f v; v.x = s; v.y = s; return v;
}

#if __has_builtin(__builtin_elementwise_fma)
__device__ __forceinline__ v2f fma2(v2f a, v2f b, v2f c) {
    return __builtin_elementwise_fma(a, b, c);   // -> v_pk_fma_f32
}
#else
__device__ __forceinline__ v2f fma2(v2f a, v2f b, v2f c) {
    v2f r;
    r.x = __builtin_fmaf(a.x, b.x, c.x);
    r.y = __builtin_fmaf(a.y, b.y, c.y);
    return r;
}
#endif

// Fast path: every thread processes exactly PTS_PER_THREAD points, no bounds
// checks (host only launches this over the multiple-of-PTS_PER_BLOCK prefix).
__global__ __launch_bounds__(THREADS) void
duffing_window_fast(const float* __restrict__ x,   // [n, 2] interleaved
                    const float* __restrict__ ws,  // [10]
                    float* __restrict__ out) {     // [n]
    // Wave-uniform weights -> SMEM loads, kept in SGPRs.
    float w[10];
#pragma unroll
    for (int k = 0; k < 10; ++k) w[k] = ws[k];

    const long long base =
        ((long long)blockIdx.x * THREADS + threadIdx.x) * PTS_PER_THREAD;
    const float* xp = x + base * 2;

    // 8 coalesced b128 loads = 16 (x0,x1) points.
    v4f L[8];
#pragma unroll
    for (int i = 0; i < 8; ++i) L[i] = *(const v4f*)(xp + 4 * i);

    // De-interleave once into 8 packed chains of 2 points each.
    v2f X0[8], X1[8];
#pragma unroll
    for (int i = 0; i < 8; ++i) {
        X0[i].x = L[i].x; X0[i].y = L[i].z;
        X1[i].x = L[i].y; X1[i].y = L[i].w;
    }

    const v2f cM01 = splat2(-0.1f);
    const v2f cP01 = splat2( 0.1f);
    const v2f cOne = splat2( 1.0f);

    // 10 steps, 8 packed chains, 5 v_pk ops per step per chain:
    //   nx0 = pk_fma(-0.1, x1, x0)
    //   nx1 = pk_fma(0.1, pk_fma(w, x1, x0*pk_fma(-x0, x0, 1)), x1)
#pragma unroll
    for (int k = 0; k < 10; ++k) {
        const v2f wk = splat2(w[k]);
#pragma unroll
        for (int j = 0; j < 8; ++j) {
            const v2f x0 = X0[j];
            const v2f x1 = X1[j];
            X0[j] = fma2(cM01, x1, x0);
            const v2f s = fma2(-x0, x0, cOne);   // 1 - x0^2
            const v2f q = fma2(wk, x1, x0 * s);  // x0 - x0^3 + w*x1
            X1[j] = fma2(cP01, q, x1);
        }
    }

    // Window: (tanh(100*(x1-0.1)) - tanh(100*(x1+0.1)) + 2) / 2
    const v2f cHundred = splat2(100.0f);
    const v2f cMTen    = splat2(-10.0f);
    const v2f cPTen    = splat2( 10.0f);
    const v2f cHalf    = splat2(  0.5f);

    v2f r[8];
#pragma unroll
    for (int j = 0; j < 8; ++j) {
        const v2f aarg = fma2(cHundred, X1[j], cMTen);
        const v2f barg = fma2(cHundred, X1[j], cPTen);
        v2f t;
        t.x = fast_tanh(aarg.x) - fast_tanh(barg.x);
        t.y = fast_tanh(aarg.y) - fast_tanh(barg.y);
        r[j] = fma2(cHalf, t, cOne);
    }

    // 4 NT b128 stores: stream the output past L2 so the input can stay
    // resident in the 192 MB L2 across graph replays.
#pragma unroll
    for (int i = 0; i < 4; ++i) {
        v4f o;
        o.x = r[2 * i].x;     o.y = r[2 * i].y;
        o.z = r[2 * i + 1].x; o.w = r[2 * i + 1].y;
        __builtin_nontemporal_store(o, (v4f*)(out + base + 4 * i));
    }
}

// Tail path: one point per thread, bounds-checked. Only launched when n is
// not a multiple of PTS_PER_BLOCK (never for the B = 16.7M harness case).
__global__ __launch_bounds__(THREADS) void
duffing_window_tail(const float* __restrict__ x,
                    const float* __restrict__ ws,
                    float* __restrict__ out,
                    int n, int n0) {
    float w[10];
#pragma unroll
    for (int k = 0; k < 10; ++k) w[k] = ws[k];

    const long long i = (long long)n0 + blockIdx.x * blockDim.x + threadIdx.x;
    if (i >= (long long)n) return;

    float x0 = x[2 * i], x1 = x[2 * i + 1];
#pragma unroll
    for (int k = 0; k < 10; ++k) {
        const float nx0 = __builtin_fmaf(-0.1f, x1, x0);
        const float s = __builtin_fmaf(-x0, x0, 1.0f);
        const float q = __builtin_fmaf(w[k], x1, x0 * s);
        x1 = __builtin_fmaf(0.1f, q, x1);
        x0 = nx0;
    }
    float ta = fast_tanh(__builtin_fmaf(100.0f, x1, -10.0f));
    float tb = fast_tanh(__builtin_fmaf(100.0f, x1,  10.0f));
    out[i] = __builtin_fmaf(0.5f, ta - tb, 1.0f);
}

extern "C" void kernel_launch(void* const* d_in, const int* in_sizes, int n_in,
                              void* d_out, int out_size, void* d_ws, size_t ws_size,
                              hipStream_t stream) {
    (void)n_in; (void)d_ws; (void)ws_size; (void)out_size;

    const float* x  = (const float*)d_in[0];   // [B, 2] float32
    const float* ws = (const float*)d_in[1];   // [10]   float32
    float* out      = (float*)d_out;           // [B]    float32

    const int n = in_sizes[0] / 2;             // number of points (B)

    const int fast_blocks = n / PTS_PER_BLOCK;             // branch-free bulk
    const int n_fast      = fast_blocks * PTS_PER_BLOCK;
    const int rem         = n - n_fast;

    if (fast_blocks > 0) {
        duffing_window_fast<<<fast_blocks, THREADS, 0, stream>>>(x, ws, out);
    }
    if (rem > 0) {
        const int tail_blocks = (rem + THREADS - 1) / THREADS;
        duffing_window_tail<<<tail_blocks, THREADS, 0, stream>>>(x, ws, out,
                                                                 n, n_fast);
    }
}